// ProteinModulationPredictor_16312285791078
// MI455X (gfx1250) — compile-verified
//
#include <hip/hip_runtime.h>
#include <hip/hip_bf16.h>

typedef __attribute__((ext_vector_type(2))) float v2f;
typedef __attribute__((ext_vector_type(8))) float v8f;

// ---------------------------------------------------------------------------
// Generic fp32 WMMA GEMM: C[M,N](ld=ldc) = act(A[M,K] @ W[K,N] + bias)
// One wave (32 threads) computes one 16x16 output tile via V_WMMA_F32_16X16X4_F32.
// M, N must be multiples of 16 (true for every GEMM in this model); K arbitrary.
// No divergence anywhere -> EXEC stays all-ones as WMMA requires.
// ---------------------------------------------------------------------------
__global__ void wmma_gemm_f32(const float* __restrict__ A, const float* __restrict__ W,
                              const float* __restrict__ bias, float* __restrict__ C,
                              int M, int K, int N, int ldc, int do_relu) {
  const int r    = blockIdx.x * 16;   // row tile
  const int c    = blockIdx.y * 16;   // col tile
  const int lane = threadIdx.x;       // 0..31
  const int half = lane >> 4;         // 0: K=k,k+1   1: K=k+2,k+3
  const int lr   = lane & 15;

  const float* __restrict__ arow = A + (long long)(r + lr) * K;
  const float* __restrict__ wcol = W + (c + lr);

  v8f acc = {};
  const int K4 = K & ~3;
  int k = 0;
  for (; k < K4; k += 4) {
    const int k0 = k + 2 * half;
    v2f a, b;
    a.x = arow[k0];
    a.y = arow[k0 + 1];
    b.x = wcol[(long long)k0 * N];
    b.y = wcol[(long long)(k0 + 1) * N];
    acc = __builtin_amdgcn_wmma_f32_16x16x4_f32(false, a, false, b, (short)0, acc,
                                                false, false);
  }
  if (k < K) {  // guarded tail for K = 9, 21
    const int k0 = k + 2 * half;
    v2f a, b;
    a.x = 0.0f; a.y = 0.0f; b.x = 0.0f; b.y = 0.0f;
    if (k0 < K)     { a.x = arow[k0];     b.x = wcol[(long long)k0 * N]; }
    if (k0 + 1 < K) { a.y = arow[k0 + 1]; b.y = wcol[(long long)(k0 + 1) * N]; }
    acc = __builtin_amdgcn_wmma_f32_16x16x4_f32(false, a, false, b, (short)0, acc,
                                                false, false);
  }

  // C/D layout: VGPR j -> row r + j + 8*half, col c + (lane&15)
  const int  col = c + lr;
  const float bv = bias ? bias[col] : 0.0f;
#pragma unroll
  for (int j = 0; j < 8; ++j) {
    const int row = r + j + 8 * half;
    float v = acc[j] + bv;
    if (do_relu) v = fmaxf(v, 0.0f);
    C[(long long)row * ldc + col] = v;
  }
}

// ---------------------------------------------------------------------------
// Small helpers
// ---------------------------------------------------------------------------
__global__ void fill_f32(float* __restrict__ p, float v, long long n) {
  long long t = (long long)blockIdx.x * blockDim.x + threadIdx.x;
  if (t < n) p[t] = v;
}

__global__ void deg_accum(const int* __restrict__ dst, float* __restrict__ deg, int E) {
  int t = blockIdx.x * blockDim.x + threadIdx.x;
  if (t < E) atomicAdd(&deg[dst[t]], 1.0f);
}

__global__ void deg_to_dinv(float* __restrict__ deg, int n) {
  int t = blockIdx.x * blockDim.x + threadIdx.x;
  if (t < n) deg[t] = rsqrtf(deg[t]);  // deg >= 1 always (self loop)
}

// scatter: out[dst] += dinv[src]*dinv[dst] * h[src];  32 lanes/edge, 4 feats/lane
__global__ void edge_scatter(const float* __restrict__ h, const int* __restrict__ src,
                             const int* __restrict__ dst, const float* __restrict__ dinv,
                             float* __restrict__ out, int E, int F) {
  const long long t = (long long)blockIdx.x * blockDim.x + threadIdx.x;
  const int lanes = F >> 2;
  const long long e = t / lanes;
  const int fi = (int)(t % lanes) << 2;
  if (e >= E) return;
  const int s = src[e], d = dst[e];
  const float coef = dinv[s] * dinv[d];
  const float4 v = *(const float4*)(h + (long long)s * F + fi);
  float* o = out + (long long)d * F + fi;
  atomicAdd(o + 0, coef * v.x);
  atomicAdd(o + 1, coef * v.y);
  atomicAdd(o + 2, coef * v.z);
  atomicAdd(o + 3, coef * v.w);
}

// agg = relu(agg + dinv[i]^2 * h + bias[f])   (self-loop term + bias + relu)
__global__ void gcn_epilogue(float* __restrict__ agg, const float* __restrict__ h,
                             const float* __restrict__ dinv, const float* __restrict__ bias,
                             int Nn, int F) {
  const long long t = (long long)blockIdx.x * blockDim.x + threadIdx.x;
  if (t >= (long long)Nn * F) return;
  const int i = (int)(t / F), f = (int)(t % F);
  const float di = dinv[i];
  agg[t] = fmaxf(agg[t] + di * di * h[t] + bias[f], 0.0f);
}

__global__ void pool_accum(const float* __restrict__ x, const int* __restrict__ batch,
                           float* __restrict__ sums, float* __restrict__ cnts,
                           int Nn, int F) {
  const long long t = (long long)blockIdx.x * blockDim.x + threadIdx.x;
  if (t >= (long long)Nn * F) return;
  const int i = (int)(t / F), f = (int)(t % F);
  const int b = batch[i];
  atomicAdd(&sums[(long long)b * F + f], x[t]);
  if (f == 0) atomicAdd(&cnts[b], 1.0f);
}

// dst[b*stride + f] = sums[b*F+f] / max(cnt[b], 1)
__global__ void pool_finish(const float* __restrict__ sums, const float* __restrict__ cnts,
                            float* __restrict__ dst, int F, int stride, int nb) {
  const int t = blockIdx.x * blockDim.x + threadIdx.x;
  if (t >= nb * F) return;
  const int b = t / F, f = t % F;
  dst[b * stride + f] = sums[t] / fmaxf(cnts[b], 1.0f);
}

// out[64,2] = relu(h[64,128] @ w[128,2] + b)
__global__ void final_out_kernel(const float* __restrict__ h, const float* __restrict__ w,
                                 const float* __restrict__ bias, float* __restrict__ out) {
  const int t = threadIdx.x;  // 128 threads
  if (t >= 128) return;
  const int b = t >> 1, o = t & 1;
  float s = bias[o];
  for (int k = 0; k < 128; ++k) s += h[b * 128 + k] * w[k * 2 + o];
  out[b * 2 + o] = fmaxf(s, 0.0f);
}

// ---------------------------------------------------------------------------
extern "C" void kernel_launch(void* const* d_in, const int* in_sizes, int n_in,
                              void* d_out, int out_size, void* d_ws, size_t ws_size,
                              hipStream_t stream) {
  const float* metadata_prot = (const float*)d_in[0];
  const float* metadata_mol  = (const float*)d_in[1];
  const float* x_prot = (const float*)d_in[2];
  const float* x_mol  = (const float*)d_in[3];
  const int* ei_prot    = (const int*)d_in[4];
  const int* batch_prot = (const int*)d_in[5];
  const int* ei_mol     = (const int*)d_in[6];
  const int* batch_mol  = (const int*)d_in[7];
  const float* fc1_w  = (const float*)d_in[8];  const float* fc1_b  = (const float*)d_in[9];
  const float* fc2_w  = (const float*)d_in[10]; const float* fc2_b  = (const float*)d_in[11];
  const float* fcm1_w = (const float*)d_in[12]; const float* fcm1_b = (const float*)d_in[13];
  const float* fcm2_w = (const float*)d_in[14]; const float* fcm2_b = (const float*)d_in[15];
  const float* gp1_w  = (const float*)d_in[16]; const float* gp1_b  = (const float*)d_in[17];
  const float* gp2_w  = (const float*)d_in[18]; const float* gp2_b  = (const float*)d_in[19];
  const float* gm1_w  = (const float*)d_in[20]; const float* gm1_b  = (const float*)d_in[21];
  const float* gm2_w  = (const float*)d_in[22]; const float* gm2_b  = (const float*)d_in[23];
  const float* fcc_w  = (const float*)d_in[24]; const float* fcc_b  = (const float*)d_in[25];
  const float* out_w  = (const float*)d_in[26]; const float* out_b  = (const float*)d_in[27];
  float* out = (float*)d_out;

  const int NP = 100000, EP = 1600000, NM = 4096, EM = 16384, NB = 64;

  // workspace carve-out (256B aligned)
  char* wsp = (char*)d_ws;
  auto carve = [&](size_t bytes) -> float* {
    float* p = (float*)wsp;
    wsp += (bytes + 255) & ~(size_t)255;
    return p;
  };
  float* bufA_p   = carve((size_t)NP * 128 * 4);
  float* bufB_p   = carve((size_t)NP * 128 * 4);
  float* dinv_p   = carve((size_t)NP * 4);
  float* bufA_m   = carve((size_t)NM * 128 * 4);
  float* bufB_m   = carve((size_t)NM * 128 * 4);
  float* dinv_m   = carve((size_t)NM * 4);
  float* mph      = carve((size_t)NB * 128 * 4);
  float* mmh      = carve((size_t)NB * 64 * 4);
  float* sums     = carve((size_t)NB * 128 * 4);
  float* cnts     = carve((size_t)NB * 4);
  float* combined = carve((size_t)NB * 448 * 4);
  float* ch       = carve((size_t)NB * 128 * 4);

  auto fillf = [&](float* p, float v, long long n) {
    fill_f32<<<(unsigned)((n + 255) / 256), 256, 0, stream>>>(p, v, n);
  };

  // ---- metadata MLPs (write final layer straight into combined cols) ----
  wmma_gemm_f32<<<dim3(NB / 16, 8), 32, 0, stream>>>(metadata_prot, fc1_w, fc1_b, mph,
                                                     NB, 18640, 128, 128, 1);
  wmma_gemm_f32<<<dim3(NB / 16, 8), 32, 0, stream>>>(mph, fc2_w, fc2_b, combined,
                                                     NB, 128, 128, 448, 1);
  wmma_gemm_f32<<<dim3(NB / 16, 4), 32, 0, stream>>>(metadata_mol, fcm1_w, fcm1_b, mmh,
                                                     NB, 21, 64, 64, 1);
  wmma_gemm_f32<<<dim3(NB / 16, 4), 32, 0, stream>>>(mmh, fcm2_w, fcm2_b, combined + 128,
                                                     NB, 64, 64, 448, 1);

  // ---- two-layer GCN + mean pool (shared pipeline) ----
  auto run_gcn = [&](const float* x, const int* ei, const int* batch, int Nn, int E,
                     const float* w1, const float* b1, const float* w2, const float* b2,
                     float* bufA, float* bufB, float* dinv, float* comb_dst) {
    const long long nf = (long long)Nn * 128;
    const long long sc = (long long)E * 32;  // scatter threads (4 feats/lane, F=128)

    fillf(dinv, 1.0f, Nn);  // self-loop
    deg_accum<<<(E + 255) / 256, 256, 0, stream>>>(ei + E, dinv, E);
    deg_to_dinv<<<(Nn + 255) / 256, 256, 0, stream>>>(dinv, Nn);

    // layer 1: h0 = x @ w1 (raw)
    wmma_gemm_f32<<<dim3(Nn / 16, 8), 32, 0, stream>>>(x, w1, nullptr, bufA,
                                                       Nn, 9, 128, 128, 0);
    fillf(bufB, 0.0f, nf);
    edge_scatter<<<(unsigned)((sc + 255) / 256), 256, 0, stream>>>(bufA, ei, ei + E,
                                                                   dinv, bufB, E, 128);
    gcn_epilogue<<<(unsigned)((nf + 255) / 256), 256, 0, stream>>>(bufB, bufA, dinv, b1,
                                                                   Nn, 128);
    // layer 2: h1 = bufB -> bufA = bufB @ w2 (raw)
    wmma_gemm_f32<<<dim3(Nn / 16, 8), 32, 0, stream>>>(bufB, w2, nullptr, bufA,
                                                       Nn, 128, 128, 128, 0);
    fillf(bufB, 0.0f, nf);
    edge_scatter<<<(unsigned)((sc + 255) / 256), 256, 0, stream>>>(bufA, ei, ei + E,
                                                                   dinv, bufB, E, 128);
    gcn_epilogue<<<(unsigned)((nf + 255) / 256), 256, 0, stream>>>(bufB, bufA, dinv, b2,
                                                                   Nn, 128);
    // mean pool -> combined columns
    fillf(sums, 0.0f, (long long)NB * 128);
    fillf(cnts, 0.0f, NB);
    pool_accum<<<(unsigned)((nf + 255) / 256), 256, 0, stream>>>(bufB, batch, sums, cnts,
                                                                 Nn, 128);
    pool_finish<<<(NB * 128 + 255) / 256, 256, 0, stream>>>(sums, cnts, comb_dst,
                                                            128, 448, NB);
  };

  run_gcn(x_prot, ei_prot, batch_prot, NP, EP, gp1_w, gp1_b, gp2_w, gp2_b,
          bufA_p, bufB_p, dinv_p, combined + 192);
  run_gcn(x_mol, ei_mol, batch_mol, NM, EM, gm1_w, gm1_b, gm2_w, gm2_b,
          bufA_m, bufB_m, dinv_m, combined + 320);

  // ---- combine head ----
  wmma_gemm_f32<<<dim3(NB / 16, 8), 32, 0, stream>>>(combined, fcc_w, fcc_b, ch,
                                                     NB, 448, 128, 128, 1);
  final_out_kernel<<<1, 128, 0, stream>>>(ch, out_w, out_b, out);
}